// UnsupervisedJazzModel_16441134809325
// MI455X (gfx1250) — compile-verified
//
#include <hip/hip_runtime.h>
#include <hip/hip_bf16.h>
#include <math.h>
#include <stdint.h>

typedef float v2f __attribute__((ext_vector_type(2)));
typedef float v8f __attribute__((ext_vector_type(8)));
typedef unsigned int u32x4 __attribute__((ext_vector_type(4)));
typedef int i32x8 __attribute__((ext_vector_type(8)));
typedef int i32x4 __attribute__((ext_vector_type(4)));

#define DD 128
#define LDW_W 132        // padded LDS row stride for W (floats)
#define LDA_A 260        // padded LDS row stride for 256-wide A block
#define LDA_O 132        // padded LDS row stride for 128-wide blocks

// ---------------------------------------------------------------------------
// WMMA f32 16x16x4 K-loop: acc += A(16xK, LDS, row stride lda) * B(KxN, row
// pitch ldw; LDS or global), one 16x16 f32 tile at column offset n0.
// A fragment (ISA 7.12.2): lane&15 = row M, half-wave picks K pair.
// B/D: lane&15 = column N, VGPR idx + 8*half = row.
// ---------------------------------------------------------------------------
__device__ __forceinline__ v8f wmma_kloop(const float* __restrict__ sA, int lda,
                                          const float* __restrict__ W, int ldw,
                                          int n0, int K, v8f acc)
{
    const int lane = threadIdx.x & 31;
    const int half = lane >> 4;
    const int m16  = lane & 15;
#pragma unroll 8
    for (int k0 = 0; k0 < K; k0 += 4) {
        const int ka = k0 + half * 2;
        v2f a, b;
        a.x = sA[m16 * lda + ka];
        a.y = sA[m16 * lda + ka + 1];
        b.x = W[(size_t)ka * ldw + n0 + m16];
        b.y = W[(size_t)(ka + 1) * ldw + n0 + m16];
        acc = __builtin_amdgcn_wmma_f32_16x16x4_f32(false, a, false, b,
                                                    (short)0, acc, false, false);
    }
    return acc;
}

// ---------------------------------------------------------------------------
// TDM: DMA one 128x128 f32 matrix (row-major, stride 128) from global into LDS
// with +4 DWORD padding every 128 DWORDs (LDS row stride becomes 132 floats).
// D# per CDNA5 ISA:
//  g0: count=1 | lds_addr[63:32] | global_addr[120:64] | type=2[127:126]
//  g1: data_size=2(4B)<<16 | pad_enable<<20 | pad_interval=6(128dw)<<22 |
//      pad_amount=3(4dw)<<25 ; tensor_dim0=128 [79:48]; tensor_dim1=128
//      [111:80]; tile_dim0=128 [127:112]; tile_dim1=128 [143:128];
//      tensor_dim0_stride=128 [207:160]
//  remaining groups: zero (2-D tensor)
// This toolchain exposes the 6-arg builtin:
//   (u32x4 g0, i32x8 g1, i32x4, i32x4, i32x8, i32 cpol)
// ---------------------------------------------------------------------------
#if __has_builtin(__builtin_amdgcn_tensor_load_to_lds) && __has_builtin(__builtin_amdgcn_s_wait_tensorcnt)
#define HAVE_TDM 1
__device__ __forceinline__ void tdm_load_128x128(const float* gW, unsigned lds_byte_off)
{
    unsigned long long ga = (unsigned long long)(uintptr_t)gW;
    u32x4 g0 = { 1u,                                   // count=1
                 lds_byte_off,                         // lds_addr
                 (unsigned)(ga & 0xffffffffu),
                 (unsigned)(((ga >> 32) & 0x1ffffffu) | (2u << 30)) };
    i32x8 g1 = { (int)((2u << 16) | (1u << 20) | (6u << 22) | (3u << 25)),
                 (int)(128u << 16),                    // tensor_dim0 lo16 @48
                 (int)(128u << 16),                    // tensor_dim1 lo16 @80
                 (int)(128u << 16),                    // tile_dim0 @112
                 128,                                  // tile_dim1 @128
                 128,                                  // tensor_dim0_stride lo32 @160
                 0, 0 };
    i32x4 z4 = { 0, 0, 0, 0 };
    i32x8 z8 = { 0, 0, 0, 0, 0, 0, 0, 0 };
    __builtin_amdgcn_tensor_load_to_lds(g0, g1, z4, z4, z8, 0);
}
#else
#define HAVE_TDM 0
#endif

// ---------------------------------------------------------------------------
// Elementwise helpers
// ---------------------------------------------------------------------------
__global__ __launch_bounds__(256) void k_zero4(float4* p, long long n4)
{
    long long i = (long long)blockIdx.x * 256 + threadIdx.x;
    if (i < n4) p[i] = make_float4(0.f, 0.f, 0.f, 0.f);
}

__global__ __launch_bounds__(256) void k_relu(float* p, long long n)
{
    long long i = (long long)blockIdx.x * 256 + threadIdx.x;
    if (i < n) p[i] = fmaxf(p[i], 0.f);
}

__global__ __launch_bounds__(256) void k_gather(const float* __restrict__ emb,
                                                const int* __restrict__ nid,
                                                float* __restrict__ dst, int n)
{
    long long i = (long long)blockIdx.x * 256 + threadIdx.x;
    long long total = (long long)n * 32;
    if (i >= total) return;
    int row = (int)(i >> 5);
    int q   = (int)(i & 31);
    int srow = nid[row];
    ((float4*)dst)[(long long)row * 32 + q] =
        ((const float4*)emb)[(long long)srow * 32 + q];
}

// one edge per wave: 32 lanes x float4 = 512B row; atomic scatter-add
__global__ __launch_bounds__(256) void k_scatter(const float* __restrict__ xsrc,
                                                 const int* __restrict__ src,
                                                 const int* __restrict__ dst,
                                                 float* __restrict__ msg,
                                                 float* __restrict__ cnt, int E)
{
    int gwave = (int)(((long long)blockIdx.x * 256 + threadIdx.x) >> 5);
    int lane  = threadIdx.x & 31;
    if (gwave >= E) return;
    int s = src[gwave];
    int d = dst[gwave];
    float4 v = ((const float4*)(xsrc + (size_t)s * DD))[lane];
    float* mr = msg + (size_t)d * DD + lane * 4;
    atomicAdd(mr + 0, v.x);
    atomicAdd(mr + 1, v.y);
    atomicAdd(mr + 2, v.z);
    atomicAdd(mr + 3, v.w);
    if (lane == 0) atomicAdd(cnt + d, 1.0f);
}

// ---------------------------------------------------------------------------
// Per-relation fused kernel:
//   out_accum[row,:] += l2norm( (msg[row,:]/max(cnt,1)) @ Wl + bl + x[row,:] @ Wr )
// Persistent blocks; [Wl;Wr] staged ONCE into LDS via TDM (256x132 padded),
// then 16-row tiles: stage A=[msg|x] in LDS, dual fp32-WMMA GEMM, bias,
// row L2-norm, accumulate. Block = 256 threads = 8 waves; wave w owns cols
// [16w,16w+16).
// Dynamic LDS layout (floats):
//   sW  [256*132]   = 33792
//   sA  [16*260]    =  4160
//   sOut[16*132]    =  2112
//   sRed[256], sInv[16]
// ---------------------------------------------------------------------------
#define SMEM_GEMM_FLOATS (33792 + 4160 + 2112 + 256 + 16)

__global__ __launch_bounds__(256) void k_gemm_rel(const float* __restrict__ msg,
                                                  const float* __restrict__ cnt,
                                                  const float* __restrict__ xdst,
                                                  const float* __restrict__ Wl,
                                                  const float* __restrict__ bl,
                                                  const float* __restrict__ Wr,
                                                  float* __restrict__ out_accum,
                                                  int n)
{
    extern __shared__ float smem[];
    float* sW   = smem;                 // 256 x 132 (Wl rows 0..127, Wr rows 128..255)
    float* sA   = sW + 33792;
    float* sOut = sA + 4160;
    float* sRed = sOut + 2112;
    float* sInv = sRed + 256;

    const int tid  = threadIdx.x;
    const int wave = tid >> 5;
    const int lane = tid & 31;
    const int half = lane >> 4;
    const int m16  = lane & 15;
    const int n0   = wave * 16;
    const int nblocks = (n + 15) >> 4;

    // ---- stage [Wl;Wr] into LDS once per (persistent) block ----
#if HAVE_TDM
    if (tid == 0) {   // TDM is per-wave, EXEC-independent: wave 0 issues both DMAs
        tdm_load_128x128(Wl, (unsigned)(size_t)sW);
        tdm_load_128x128(Wr, (unsigned)(size_t)(sW + 128 * LDW_W));
        __builtin_amdgcn_s_wait_tensorcnt(0);
    }
#else
    for (int idx = tid; idx < 256 * DD; idx += 256) {
        int k = idx >> 7, c = idx & 127;
        sW[k * LDW_W + c] = (k < DD) ? Wl[idx] : Wr[(size_t)(k - DD) * DD + c];
    }
#endif
    __syncthreads();

    for (int blk = blockIdx.x; blk < nblocks; blk += gridDim.x) {
        const int row0 = blk << 4;
        // ---- stage A = [msg/cnt | x] : 16 rows x 256, coalesced ----
        for (int idx = tid; idx < 16 * 256; idx += 256) {
            int r = idx >> 8;
            int k = idx & 255;
            int gr = row0 + r;
            float v = 0.f;
            if (gr < n) {
                if (k < DD) {
                    float inv = 1.0f / fmaxf(cnt[gr], 1.0f);
                    v = msg[(size_t)gr * DD + k] * inv;
                } else {
                    v = xdst[(size_t)gr * DD + (k - DD)];
                }
            }
            sA[r * LDA_A + k] = v;
        }
        __syncthreads();

        // ---- dual GEMM on the matrix pipe (fp32 WMMA), B from LDS ----
        v8f acc = {};
        acc = wmma_kloop(sA, LDA_A, sW, LDW_W, n0, DD, acc);                    // msg @ Wl
        acc = wmma_kloop(sA + DD, LDA_A, sW + 128 * LDW_W, LDW_W, n0, DD, acc); // + x @ Wr

        // ---- bias, stash D tile to LDS ----
        float bias = bl[n0 + m16];
#pragma unroll
        for (int i = 0; i < 8; ++i) {
            int mm = i + 8 * half;
            sOut[mm * LDA_O + n0 + m16] = acc[i] + bias;
        }
        __syncthreads();

        // ---- per-row L2 norm ----
        {
            int r = tid >> 4, c = tid & 15;
            float s = 0.f;
#pragma unroll
            for (int j = 0; j < 8; ++j) {
                float v = sOut[r * LDA_O + c + 16 * j];
                s += v * v;
            }
            sRed[r * 16 + c] = s;
        }
        __syncthreads();
        if (tid < 16) {
            float s = 0.f;
            for (int c = 0; c < 16; ++c) s += sRed[tid * 16 + c];
            sInv[tid] = 1.0f / fmaxf(sqrtf(s), 1e-12f);
        }
        __syncthreads();

        // ---- accumulate into new[t] (launch-serialized, no atomics needed) ----
        for (int idx = tid; idx < 16 * DD; idx += 256) {
            int r = idx >> 7, c = idx & 127;
            int gr = row0 + r;
            if (gr < n)
                out_accum[(size_t)gr * DD + c] += sOut[r * LDA_O + c] * sInv[r];
        }
        __syncthreads();
    }
}

// ---------------------------------------------------------------------------
// Head: out[row,:] = l2norm( relu( LN(x[row,:]) @ pw1 + pb1 ) @ pw2 + pb2 )
// ---------------------------------------------------------------------------
__global__ __launch_bounds__(256) void k_final(const float* __restrict__ x,
                                               const float* __restrict__ lng,
                                               const float* __restrict__ lnb,
                                               const float* __restrict__ pw1,
                                               const float* __restrict__ pb1,
                                               const float* __restrict__ pw2,
                                               const float* __restrict__ pb2,
                                               float* __restrict__ out, int n)
{
    __shared__ float sA[16 * LDA_O];
    __shared__ float sOut[16 * LDA_O];
    __shared__ float sRed[16 * 16];
    __shared__ float sRed2[16 * 16];
    __shared__ float sMu[16];
    __shared__ float sIs[16];
    __shared__ float sInv[16];

    const int tid  = threadIdx.x;
    const int wave = tid >> 5;
    const int lane = tid & 31;
    const int half = lane >> 4;
    const int m16  = lane & 15;
    const int n0   = wave * 16;
    const int nblocks = (n + 15) >> 4;

    for (int blk = blockIdx.x; blk < nblocks; blk += gridDim.x) {
        const int row0 = blk << 4;
        // stage raw x
        for (int idx = tid; idx < 16 * DD; idx += 256) {
            int r = idx >> 7, c = idx & 127;
            int gr = row0 + r;
            sA[r * LDA_O + c] = (gr < n) ? x[(size_t)gr * DD + c] : 0.f;
        }
        __syncthreads();
        // row stats
        {
            int r = tid >> 4, c = tid & 15;
            float s1 = 0.f, s2 = 0.f;
#pragma unroll
            for (int j = 0; j < 8; ++j) {
                float v = sA[r * LDA_O + c + 16 * j];
                s1 += v; s2 += v * v;
            }
            sRed[r * 16 + c] = s1;
            sRed2[r * 16 + c] = s2;
        }
        __syncthreads();
        if (tid < 16) {
            float s1 = 0.f, s2 = 0.f;
            for (int c = 0; c < 16; ++c) { s1 += sRed[tid * 16 + c]; s2 += sRed2[tid * 16 + c]; }
            float mu = s1 * (1.0f / DD);
            float var = fmaxf(s2 * (1.0f / DD) - mu * mu, 0.f);
            sMu[tid] = mu;
            sIs[tid] = rsqrtf(var + 1e-5f);
        }
        __syncthreads();
        // layernorm in place
        for (int idx = tid; idx < 16 * DD; idx += 256) {
            int r = idx >> 7, c = idx & 127;
            float v = sA[r * LDA_O + c];
            sA[r * LDA_O + c] = (v - sMu[r]) * sIs[r] * lng[c] + lnb[c];
        }
        __syncthreads();

        // GEMM1 (WMMA), relu, back into sA
        v8f acc = {};
        acc = wmma_kloop(sA, LDA_O, pw1, DD, n0, DD, acc);
        __syncthreads();   // everyone done reading sA before overwrite
        {
            float b1 = pb1[n0 + m16];
#pragma unroll
            for (int i = 0; i < 8; ++i) {
                int mm = i + 8 * half;
                sA[mm * LDA_O + n0 + m16] = fmaxf(acc[i] + b1, 0.f);
            }
        }
        __syncthreads();

        // GEMM2 (WMMA) + bias
        v8f acc2 = {};
        acc2 = wmma_kloop(sA, LDA_O, pw2, DD, n0, DD, acc2);
        {
            float b2 = pb2[n0 + m16];
#pragma unroll
            for (int i = 0; i < 8; ++i) {
                int mm = i + 8 * half;
                sOut[mm * LDA_O + n0 + m16] = acc2[i] + b2;
            }
        }
        __syncthreads();

        // L2 norm + write
        {
            int r = tid >> 4, c = tid & 15;
            float s = 0.f;
#pragma unroll
            for (int j = 0; j < 8; ++j) {
                float v = sOut[r * LDA_O + c + 16 * j];
                s += v * v;
            }
            sRed[r * 16 + c] = s;
        }
        __syncthreads();
        if (tid < 16) {
            float s = 0.f;
            for (int c = 0; c < 16; ++c) s += sRed[tid * 16 + c];
            sInv[tid] = 1.0f / fmaxf(sqrtf(s), 1e-12f);
        }
        __syncthreads();
        for (int idx = tid; idx < 16 * DD; idx += 256) {
            int r = idx >> 7, c = idx & 127;
            int gr = row0 + r;
            if (gr < n)
                out[(size_t)gr * DD + c] = sOut[r * LDA_O + c] * sInv[r];
        }
        __syncthreads();
    }
}

// ---------------------------------------------------------------------------
// Host orchestration
// ---------------------------------------------------------------------------
static inline void launch_zero(float* p, long long nfloats, hipStream_t s)
{
    long long n4 = nfloats >> 2;   // all our sizes are multiples of 4
    int grid = (int)((n4 + 255) / 256);
    hipLaunchKernelGGL(k_zero4, dim3(grid), dim3(256), 0, s, (float4*)p, n4);
}

extern "C" void kernel_launch(void* const* d_in, const int* in_sizes, int n_in,
                              void* d_out, int out_size, void* d_ws, size_t ws_size,
                              hipStream_t stream)
{
    (void)n_in; (void)out_size; (void)ws_size;
    const int NP = 200000, NA = 30000, NS = 50000;
    const int NN[3] = {NP, NA, NS};                // perf, artist, song
    const int L = 3, R = 6;

    const int* nid[3] = {(const int*)d_in[0], (const int*)d_in[1], (const int*)d_in[2]};
    const float* emb[3] = {(const float*)d_in[9], (const float*)d_in[10], (const float*)d_in[11]};
    const float* Wl = (const float*)d_in[12];
    const float* bl = (const float*)d_in[13];
    const float* Wr = (const float*)d_in[14];
    const float* lng = (const float*)d_in[15];
    const float* lnb = (const float*)d_in[16];
    const float* pw1 = (const float*)d_in[17];
    const float* pb1 = (const float*)d_in[18];
    const float* pw2 = (const float*)d_in[19];
    const float* pb2 = (const float*)d_in[20];

    // relation topology: (src type, dst type), type order {perf,artist,song}
    const int rel_src[R] = {1, 0, 1, 0, 2, 2};
    const int rel_dst[R] = {0, 2, 2, 1, 0, 1};

    // workspace layout (floats)
    float* ws = (float*)d_ws;
    size_t off = 0;
    float* xbuf[2][3];
    for (int s = 0; s < 2; ++s)
        for (int t = 0; t < 3; ++t) { xbuf[s][t] = ws + off; off += (size_t)NN[t] * DD; }
    float* msg = ws + off; off += (size_t)NP * DD;   // sized for largest dst
    float* cnt = ws + off; off += (size_t)NP;

    float* xcur[3] = {xbuf[0][0], xbuf[0][1], xbuf[0][2]};
    float* xnxt[3] = {xbuf[1][0], xbuf[1][1], xbuf[1][2]};

    // initial gather x[t] = emb[t][nid]
    for (int t = 0; t < 3; ++t) {
        long long tot = (long long)NN[t] * 32;
        int grid = (int)((tot + 255) / 256);
        hipLaunchKernelGGL(k_gather, dim3(grid), dim3(256), 0, stream,
                           emb[t], nid[t], xcur[t], NN[t]);
    }

    const size_t smemGemm = (size_t)SMEM_GEMM_FLOATS * sizeof(float);

    for (int l = 0; l < L; ++l) {
        for (int t = 0; t < 3; ++t)
            launch_zero(xnxt[t], (long long)NN[t] * DD, stream);

        for (int r = 0; r < R; ++r) {
            int st = rel_src[r], dt = rel_dst[r];
            int ndst = NN[dt];
            const int* eptr = (const int*)d_in[3 + r];
            int E = in_sizes[3 + r] / 2;
            const int* esrc = eptr;
            const int* edst = eptr + E;

            launch_zero(msg, (long long)ndst * DD, stream);
            launch_zero(cnt, (long long)ndst, stream);

            int sgrid = (E + 7) / 8;   // 8 edges per 256-thread block (1 edge/wave)
            hipLaunchKernelGGL(k_scatter, dim3(sgrid), dim3(256), 0, stream,
                               xcur[st], esrc, edst, msg, cnt, E);

            const float* wlp = Wl + ((size_t)(l * R + r)) * DD * DD;
            const float* wrp = Wr + ((size_t)(l * R + r)) * DD * DD;
            const float* blp = bl + ((size_t)(l * R + r)) * DD;
            int nblk = (ndst + 15) / 16;
            int ggrid = nblk < 2048 ? nblk : 2048;
            hipLaunchKernelGGL(k_gemm_rel, dim3(ggrid), dim3(256), smemGemm, stream,
                               msg, cnt, xcur[dt], wlp, blp, wrp, xnxt[dt], ndst);
        }

        if (l < L - 1) {
            for (int t = 0; t < 3; ++t) {
                long long tot = (long long)NN[t] * DD;
                int grid = (int)((tot + 255) / 256);
                hipLaunchKernelGGL(k_relu, dim3(grid), dim3(256), 0, stream,
                                   xnxt[t], tot);
            }
        }
        for (int t = 0; t < 3; ++t) { float* tmp = xcur[t]; xcur[t] = xnxt[t]; xnxt[t] = tmp; }
    }

    // heads; output order: performance, artist, song (concatenated flat)
    float* out = (float*)d_out;
    size_t ooff = 0;
    for (int t = 0; t < 3; ++t) {
        int nblk = (NN[t] + 15) / 16;
        int ggrid = nblk < 2048 ? nblk : 2048;
        hipLaunchKernelGGL(k_final, dim3(ggrid), dim3(256), 0, stream,
                           xcur[t],
                           lng + (size_t)t * DD, lnb + (size_t)t * DD,
                           pw1 + (size_t)t * DD * DD, pb1 + (size_t)t * DD,
                           pw2 + (size_t)t * DD * DD, pb2 + (size_t)t * DD,
                           out + ooff, NN[t]);
        ooff += (size_t)NN[t] * DD;
    }
}